// DifferentiableChamferLoss_84155589197954
// MI455X (gfx1250) — compile-verified
//
#include <hip/hip_runtime.h>
#include <hip/hip_bf16.h>

typedef __attribute__((ext_vector_type(2))) float v2f;
typedef __attribute__((ext_vector_type(8))) float v8f;

#define FOCAL_UM  5000.0f
#define PITCH_UM  150.0f
#define GRIDSZ    128
#define SENSOR_W  (GRIDSZ * PITCH_UM)   /* 19200 */
#define SENSOR_H  (GRIDSZ * PITCH_UM)
#define N_SUB     (GRIDSZ * GRIDSZ)     /* 16384 */
#define NTILES    (N_SUB / 16)          /* 1024 n-tiles per batch */
#define BB        8
#define MM        512
#define CAPV      5.0f
#define LARGE_D2  1.0e12f               /* sqrt/PITCH ~ 6.7e3 >> CAP, same post-cap result as BIG */
#define NSPLIT    8
#define NT_PER_SPLIT (NTILES / NSPLIT)  /* 128 n-tiles per wave */

// ---------------------------------------------------------------------------
// Kernel 0: init accumulators (min buffer seeded with +FLT_MAX bit pattern)
// ---------------------------------------------------------------------------
__global__ void init_kernel(unsigned* __restrict__ min_d2,
                            float* __restrict__ fb_sum,
                            int* __restrict__ n_ib) {
    int i = blockIdx.x * blockDim.x + threadIdx.x;
    if (i < BB * MM) min_d2[i] = 0x7F7FFFFFu;   // FLT_MAX as uint (d2 >= 0 ordering)
    if (i < BB) { fb_sum[i] = 0.0f; n_ib[i] = 0; }
}

// ---------------------------------------------------------------------------
// Kernel 1: slopes -> E -> A matrix pre-swizzled into WMMA lane layout.
//   Per 16-row tile: 32 consecutive float2 entries,
//     entry r       (lanes 0-15):  (Ex, Ey)   = K0,K1 of row r   [(0,0) if masked]
//     entry r + 16  (lanes 16-31): (aug, 1.0) = K2,K3 of row r   [aug = e2 or LARGE]
//   Also accumulates fb_sum / n_ib with wave-reduced atomics.
// ---------------------------------------------------------------------------
__global__ void setup_kernel(const float* __restrict__ pred,   // (B,9)
                             const float* __restrict__ G,      // (2N,10) row-major
                             const float* __restrict__ ref,    // (N,2)
                             float2* __restrict__ Aw,          // (B, NTILES, 32)
                             float* __restrict__ fb_sum,
                             int* __restrict__ n_ib) {
    int idx = blockIdx.x * blockDim.x + threadIdx.x;
    if (idx >= BB * N_SUB) return;
    int b = idx / N_SUB;
    int n = idx % N_SUB;

    const float* pc = pred + b * 9;
    const float* gx = G + (size_t)n * 10;
    const float* gy = G + (size_t)(N_SUB + n) * 10;
    float sx = 0.0f, sy = 0.0f;
    #pragma unroll
    for (int k = 0; k < 9; ++k) {          // full[b][0] == 0 -> skip G[:,0]
        float c = pc[k];
        sx = fmaf(c, gx[k + 1], sx);
        sy = fmaf(c, gy[k + 1], sy);
    }
    float Ex = ref[2 * n + 0] + FOCAL_UM * sx;
    float Ey = ref[2 * n + 1] + FOCAL_UM * sy;

    const float margin = 0.5f * PITCH_UM;
    bool inb = (Ex >= -margin) && (Ex <= SENSOR_W + margin) &&
               (Ey >= -margin) && (Ey <= SENSOR_H + margin);
    float e2 = fmaf(Ex, Ex, Ey * Ey);

    int nt = n >> 4;          // tile within batch
    int r  = n & 15;          // row within tile
    float2* tile = Aw + (((size_t)b * NTILES + nt) << 5);
    tile[r]      = inb ? make_float2(Ex, Ey) : make_float2(0.0f, 0.0f);
    tile[r + 16] = make_float2(inb ? e2 : LARGE_D2, 1.0f);

    // fallback term uses unmasked E
    float dx = Ex - 0.5f * SENSOR_W, dy = Ey - 0.5f * SENSOR_H;
    float rr = sqrtf(fmaf(dx, dx, dy * dy));

    // wave32 reduction, one atomic per wave (block never spans batches: 16384 % 256 == 0)
    unsigned long long bal = __ballot(inb ? 1 : 0);
    int cnt = __popcll(bal);
    #pragma unroll
    for (int off = 16; off > 0; off >>= 1) rr += __shfl_down(rr, off);
    if ((threadIdx.x & 31) == 0) {
        atomicAdd(&fb_sum[b], rr);
        atomicAdd(&n_ib[b], cnt);
    }
}

// ---------------------------------------------------------------------------
// Kernel 2: pairwise d2 via V_WMMA_F32_16X16X4_F32 (augmented K=4 trick).
//   A operand loads straight from the pre-swizzled layout (1 b64/lane/tile,
//   zero lane-selects in the loop). Two alternating running-min accumulators
//   keep the v_min dependency chain short so VALU co-issues with WMMA.
// ---------------------------------------------------------------------------
__global__ void __launch_bounds__(32)
chamfer_wmma_kernel(const float* __restrict__ observed, // (B,M,2)
                    const float2* __restrict__ Aw,      // (B, NTILES, 32)
                    unsigned* __restrict__ min_d2) {    // (B,M)
    int wid = blockIdx.x;                       // B * (M/16) * NSPLIT waves
    int s   = wid % NSPLIT;
    int mt  = (wid / NSPLIT) % (MM / 16);
    int b   = wid / (NSPLIT * (MM / 16));

    int lane = threadIdx.x;
    int c    = lane & 15;
    int hi   = lane >> 4;

    // B matrix 4x16: col c = [-2Ox, -2Oy, 1, o2]; lanes<16 hold K=0,1; lanes>=16 hold K=2,3
    int m = mt * 16 + c;
    float Ox = observed[((size_t)b * MM + m) * 2 + 0];
    float Oy = observed[((size_t)b * MM + m) * 2 + 1];
    float o2 = fmaf(Ox, Ox, Oy * Oy);
    v2f bmat;
    bmat.x = hi ? 1.0f : (-2.0f * Ox);
    bmat.y = hi ? o2   : (-2.0f * Oy);

    const float2* Ab = Aw + ((((size_t)b * NTILES) + (size_t)s * NT_PER_SPLIT) << 5) + lane;

    float lmin0 = 3.0e38f, lmin1 = 3.0e38f;
    for (int nt = 0; nt < NT_PER_SPLIT; nt += 2) {
        // --- tile nt ---
        float2 a0 = Ab[(size_t)nt << 5];
        v2f am0; am0.x = a0.x; am0.y = a0.y;
        v8f acc0 = {};
        acc0 = __builtin_amdgcn_wmma_f32_16x16x4_f32(
            false, am0, false, bmat, (short)0, acc0, false, false);
        float t00 = fminf(fminf(acc0[0], acc0[1]), fminf(acc0[2], acc0[3]));
        float t01 = fminf(fminf(acc0[4], acc0[5]), fminf(acc0[6], acc0[7]));
        lmin0 = fminf(lmin0, fminf(t00, t01));

        // --- tile nt+1 (independent accumulator) ---
        float2 a1 = Ab[(size_t)(nt + 1) << 5];
        v2f am1; am1.x = a1.x; am1.y = a1.y;
        v8f acc1 = {};
        acc1 = __builtin_amdgcn_wmma_f32_16x16x4_f32(
            false, am1, false, bmat, (short)0, acc1, false, false);
        float t10 = fminf(fminf(acc1[0], acc1[1]), fminf(acc1[2], acc1[3]));
        float t11 = fminf(fminf(acc1[4], acc1[5]), fminf(acc1[6], acc1[7]));
        lmin1 = fminf(lmin1, fminf(t10, t11));
    }
    float lmin = fminf(lmin0, lmin1);

    // combine the two lane halves (rows 0-7 vs 8-15 of each tile)
    lmin = fminf(lmin, __shfl_xor(lmin, 16));
    if (hi == 0) {
        unsigned bits = __float_as_uint(fmaxf(lmin, 0.0f)); // d2 >= 0 -> uint order == float order
        atomicMin(&min_d2[(size_t)b * MM + m], bits);
    }
}

// ---------------------------------------------------------------------------
// Kernel 3: sqrt / cap / means / fallback -> scalar
// ---------------------------------------------------------------------------
__global__ void finalize_kernel(const unsigned* __restrict__ min_d2,
                                const float* __restrict__ fb_sum,
                                const int* __restrict__ n_ib,
                                float* __restrict__ out) {
    __shared__ float red[512];
    int t = threadIdx.x;
    float total = 0.0f;
    for (int b = 0; b < BB; ++b) {
        float d2 = __uint_as_float(min_d2[b * MM + t]);
        float dist = sqrtf(d2) * (1.0f / PITCH_UM);
        red[t] = fminf(dist, CAPV);
        __syncthreads();
        for (int off = 256; off > 0; off >>= 1) {
            if (t < off) red[t] += red[t + off];
            __syncthreads();
        }
        if (t == 0) {
            float loss_main = red[0] * (1.0f / (float)MM);
            float fb = fb_sum[b] * (1.0f / (float)N_SUB) * (1.0f / PITCH_UM) + 10.0f;
            total += (n_ib[b] < 5) ? fb : loss_main;
        }
        __syncthreads();
    }
    if (t == 0) out[0] = total * (1.0f / (float)BB);
}

// ---------------------------------------------------------------------------
extern "C" void kernel_launch(void* const* d_in, const int* in_sizes, int n_in,
                              void* d_out, int out_size, void* d_ws, size_t ws_size,
                              hipStream_t stream) {
    const float* pred = (const float*)d_in[0];   // (B,9)
    const float* obs  = (const float*)d_in[1];   // (B,M,2)
    const float* G    = (const float*)d_in[2];   // (2N,10)
    const float* ref  = (const float*)d_in[3];   // (N,2)
    float* out = (float*)d_out;

    char* ws = (char*)d_ws;
    float2*   Aw     = (float2*)ws;                                        // 2 MB
    unsigned* min_d2 = (unsigned*)(ws + sizeof(float2) * 2 * BB * N_SUB);  // 16 KB
    float*    fb_sum = (float*)((char*)min_d2 + sizeof(unsigned) * BB * MM);
    int*      nib    = (int*)((char*)fb_sum + sizeof(float) * BB);

    init_kernel<<<(BB * MM + 255) / 256, 256, 0, stream>>>(min_d2, fb_sum, nib);
    setup_kernel<<<(BB * N_SUB) / 256, 256, 0, stream>>>(pred, G, ref, Aw, fb_sum, nib);
    chamfer_wmma_kernel<<<BB * (MM / 16) * NSPLIT, 32, 0, stream>>>(obs, Aw, min_d2);
    finalize_kernel<<<1, 512, 0, stream>>>(min_d2, fb_sum, nib, out);
}